// ToBoxes_52158082843018
// MI455X (gfx1250) — compile-verified
//
#include <hip/hip_runtime.h>
#include <cstdint>

// ToBoxes for MI455X (gfx1250): 3x3 maxpool stencil + masked 5-channel pack.
// Bandwidth-bound (~302 MB total => ~13us @ 23.3 TB/s). Strategy:
//  - stage heatmap halo tiles (18x66) in LDS via gfx1250 async global->LDS
//    (ASYNCcnt / s_wait_asynccnt), zero-filling out-of-image halo,
//  - regular (RT) caching for heatmap (halo rows shared between tiles via L2),
//  - non-temporal loads for read-once `sizes`, non-temporal b128 stores for
//    the 201MB write-once output,
//  - wave32-friendly 256-thread blocks, 4 pixels/thread, all b128 accesses
//    16B-aligned.

typedef float v4f __attribute__((ext_vector_type(4)));
typedef int   v4i __attribute__((vector_size(16)));

#define IMG        512
#define TS_X       16            // tile rows (x / dim-W)
#define TS_Y       64            // tile cols (y / dim-H, contiguous)
#define LDS_STRIDE 72            // floats per halo row (16B-aligned vec slots)
#define THRESH     0.3f

#if defined(__HIP_DEVICE_COMPILE__) && __has_builtin(__builtin_amdgcn_global_load_async_to_lds_b128)
#define USE_ASYNC_LDS 1
typedef __attribute__((address_space(1))) v4i* gv4i_p;   // global (AS1)
typedef __attribute__((address_space(3))) v4i* lv4i_p;   // LDS (AS3)
#else
#define USE_ASYNC_LDS 0
#endif

__global__ __launch_bounds__(256) void ToBoxes_kernel(
    const float* __restrict__ hm,   // (32,1,512,512)
    const float* __restrict__ sz,   // (32,2,512,512)
    float* __restrict__ okp,        // (32,512,512) as 0.0/1.0
    float* __restrict__ o5)         // (32,512,512,5)
{
    // halo tile: rows r=0..17 <-> gx = x0-1+r ; cols c <-> gy = y0 + (c-4)
    // valid c range [3,68]; c=4..67 filled by 16B async vectors, c=3/68 scalar.
    __shared__ float tile[18 * LDS_STRIDE];

    const int tid = threadIdx.x;
    const int y0  = blockIdx.x * TS_Y;
    const int x0  = blockIdx.y * TS_X;
    const int b   = blockIdx.z;
    const float* hmb = hm + (size_t)b * (IMG * IMG);

    // ---- stage interior + row-halo: 18 rows x 16 float4 each ----
    #pragma unroll 2
    for (int idx = tid; idx < 18 * 16; idx += 256) {
        const int r  = idx >> 4;
        const int v  = idx & 15;
        const int gx = x0 - 1 + r;
        float* lp = &tile[r * LDS_STRIDE + 4 + 4 * v];
        if ((unsigned)gx < (unsigned)IMG) {
            const float* gp = hmb + (size_t)gx * IMG + (y0 + 4 * v);
#if USE_ASYNC_LDS
            __builtin_amdgcn_global_load_async_to_lds_b128(
                (gv4i_p)(uintptr_t)gp,
                (lv4i_p)lp,
                /*offset=*/0, /*cpol=*/0);
#else
            *(v4f*)lp = *(const v4f*)gp;
#endif
        } else {
            v4f z = {0.f, 0.f, 0.f, 0.f};   // zero halo: safe, inputs >= 0
            *(v4f*)lp = z;
        }
    }

    // ---- left/right halo columns: 18 rows x 2 scalars ----
    if (tid < 36) {
        const int r    = tid >> 1;
        const int side = tid & 1;
        const int gx   = x0 - 1 + r;
        const int gy   = side ? (y0 + TS_Y) : (y0 - 1);
        const int c    = side ? (TS_Y + 4) : 3;
        float val = 0.f;
        if ((unsigned)gx < (unsigned)IMG && (unsigned)gy < (unsigned)IMG)
            val = hmb[(size_t)gx * IMG + gy];
        tile[r * LDS_STRIDE + c] = val;
    }

#if USE_ASYNC_LDS
#if __has_builtin(__builtin_amdgcn_s_wait_asynccnt)
    __builtin_amdgcn_s_wait_asynccnt(0);
#else
    asm volatile("s_wait_asynccnt 0" ::: "memory");
#endif
#endif
    __syncthreads();

    // ---- compute: each thread -> 4 consecutive y pixels of one x row ----
    const int ox  = tid >> 4;            // 0..15
    const int oyc = (tid & 15) << 2;     // 0,4,...,60
    const float* t0 = &tile[(ox + 0) * LDS_STRIDE + oyc + 3];
    const float* t1 = &tile[(ox + 1) * LDS_STRIDE + oyc + 3];
    const float* t2 = &tile[(ox + 2) * LDS_STRIDE + oyc + 3];

    float mid[6], cm[6];
    #pragma unroll
    for (int c2 = 0; c2 < 6; ++c2) {
        const float a = t0[c2], m = t1[c2], d = t2[c2];
        mid[c2] = m;
        cm[c2]  = fmaxf(fmaxf(a, m), d);
    }

    const int x  = x0 + ox;
    const int yb = y0 + oyc;
    const float cx = (float)x * (1.0f / IMG);

    // read-once sizes: non-temporal 128b loads
    const size_t szb = (size_t)b * (2 * IMG * IMG) + (size_t)x * IMG + yb;
    const v4f w4 = __builtin_nontemporal_load((const v4f*)(sz + szb));
    const v4f h4 = __builtin_nontemporal_load((const v4f*)(sz + szb + IMG * IMG));

    v4f kpv;
    v4f ov[5];
    float* o = (float*)ov;
    #pragma unroll
    for (int k = 0; k < 4; ++k) {
        const float h      = mid[k + 1];
        const float pooled = fmaxf(fmaxf(cm[k], cm[k + 1]), cm[k + 2]);
        const bool  kp     = (pooled == h) && (h > THRESH);
        kpv[k] = kp ? 1.0f : 0.0f;
        const float cy = (float)(yb + k) * (1.0f / IMG);
        o[5 * k + 0] = kp ? h     : 0.f;
        o[5 * k + 1] = kp ? cx    : 0.f;
        o[5 * k + 2] = kp ? cy    : 0.f;
        o[5 * k + 3] = kp ? w4[k] : 0.f;
        o[5 * k + 4] = kp ? h4[k] : 0.f;
    }

    // ---- write-once streaming output: non-temporal b128 stores ----
    const size_t kidx = (size_t)b * (IMG * IMG) + (size_t)x * IMG + yb;
    __builtin_nontemporal_store(kpv, (v4f*)(okp + kidx));
    float* o5p = o5 + kidx * 5;          // yb % 4 == 0 -> 80B-aligned
    #pragma unroll
    for (int q = 0; q < 5; ++q)
        __builtin_nontemporal_store(ov[q], (v4f*)(o5p + 4 * q));
}

extern "C" void kernel_launch(void* const* d_in, const int* in_sizes, int n_in,
                              void* d_out, int out_size, void* d_ws, size_t ws_size,
                              hipStream_t stream) {
    (void)in_sizes; (void)n_in; (void)out_size; (void)d_ws; (void)ws_size;
    const float* hm = (const float*)d_in[0];            // 32*1*512*512
    const float* sz = (const float*)d_in[1];            // 32*2*512*512
    float* out    = (float*)d_out;
    float* out_kp = out;                                // 8,388,608 floats (bool as 0/1)
    float* out5   = out + (size_t)32 * IMG * IMG;       // 41,943,040 floats

    dim3 grid(IMG / TS_Y, IMG / TS_X, 32);              // (8, 32, 32)
    ToBoxes_kernel<<<grid, 256, 0, stream>>>(hm, sz, out_kp, out5);
}